// Top_k_features_68023692034558
// MI455X (gfx1250) — compile-verified
//
#include <hip/hip_runtime.h>
#include <hip/hip_bf16.h>
#include <stdint.h>

// Top-k(8) of adj[n,m]*x[n,f] over n, per (m,f); out[m,0,f]=x[m,f].
// N=2048 nodes, F=64 features, K=8. Output [2048, 9, 64] f32.
//
// Streaming-selection kernel (no contractable FLOPs -> WMMA inapplicable).
// Uses CDNA5 async global->LDS DMA (ASYNCcnt) with double buffering.

#define N_NODES 2048
#define N_FEAT  64
#define K_TOP   8
#define CN      64                  // n-chunk staged per double-buffer phase
#define NCHUNK  (N_NODES / CN)      // 32

// One async b128: each of the 32 lanes moves 16 contiguous bytes
// from global memory at byte offset goff past base to the LDS address lds_off. Tracked by ASYNCcnt.
__device__ __forceinline__ void async_b128(uint32_t lds_off, uint32_t goff,
                                           const float* base) {
  asm volatile("global_load_async_to_lds_b128 %0, %1, %2 offset:0"
               :: "v"(lds_off), "v"(goff), "s"(base)
               : "memory");
}

__device__ __forceinline__ void wait_async0() {
  asm volatile("s_wait_asynccnt 0" ::: "memory");
}

// Load a CN x 32 float tile (rows of 128B) into LDS. One wave, 16 b128 ops.
// lane l covers row (4*i + l/8), bytes (l%8)*16 within the row.
__device__ __forceinline__ void load_tile(uint32_t lds_base, const float* base,
                                          uint32_t row0, uint32_t row_stride_elts,
                                          uint32_t col0, int lane) {
  const uint32_t lrow = (uint32_t)lane >> 3;
  const uint32_t lcol = ((uint32_t)lane & 7u) * 16u;
#pragma unroll
  for (int i = 0; i < 16; ++i) {
    const uint32_t r = (uint32_t)i * 4u + lrow;
    async_b128(lds_base + r * 128u + lcol,
               ((row0 + r) * row_stride_elts + col0) * 4u + lcol,
               base);
  }
}

__global__ __launch_bounds__(1024)
void topk_feat_kernel(const float* __restrict__ x,
                      const float* __restrict__ adj,
                      float* __restrict__ out) {
  __shared__ float s_adj[2][CN][32];   // [buf][n'][m_local]  8KB/buf
  __shared__ float s_x  [2][CN][32];   // [buf][n'][f_local]  8KB/buf
  __shared__ float s_o  [32][33];      // padded transpose staging

  const int tx = threadIdx.x;          // m_local = lane (wave32)
  const int ty = threadIdx.y;          // f_local = wave id
  const int m0 = blockIdx.x * 32;
  const int f0 = blockIdx.y * 32;

  const uint32_t adj_lds[2] = { (uint32_t)(uintptr_t)&s_adj[0][0][0],
                                (uint32_t)(uintptr_t)&s_adj[1][0][0] };
  const uint32_t x_lds[2]   = { (uint32_t)(uintptr_t)&s_x[0][0][0],
                                (uint32_t)(uintptr_t)&s_x[1][0][0] };

  // ---- prologue: chunk 0 -> buffer 0 (wave0: adj tile, wave1: x tile)
  if (ty == 0)      load_tile(adj_lds[0], adj, 0u, N_NODES, (uint32_t)m0, tx);
  else if (ty == 1) load_tile(x_lds[0],   x,   0u, N_FEAT,  (uint32_t)f0, tx);

  const float NEG = -__builtin_inff();
  float t0 = NEG, t1 = NEG, t2 = NEG, t3 = NEG,
        t4 = NEG, t5 = NEG, t6 = NEG, t7 = NEG;

  for (int c = 0; c < NCHUNK; ++c) {
    wait_async0();           // issuing waves: chunk c landed in LDS
    __syncthreads();         // everyone past compute(c-1); data(c) visible

    // issue chunk c+1 into the buffer freed by compute(c-1)
    if (c + 1 < NCHUNK) {
      const int nb = (c + 1) & 1;
      const uint32_t n0 = (uint32_t)(c + 1) * CN;
      if (ty == 0)      load_tile(adj_lds[nb], adj, n0, N_NODES, (uint32_t)m0, tx);
      else if (ty == 1) load_tile(x_lds[nb],   x,   n0, N_FEAT,  (uint32_t)f0, tx);
    }

    // ---- selection over this chunk
    const int cb = c & 1;
    const float* __restrict__ pa = &s_adj[cb][0][tx];  // stride 32, bank-clean
    const float* __restrict__ px = &s_x[cb][0][ty];    // wave-uniform broadcast
#pragma unroll 8
    for (int n = 0; n < CN; ++n) {
      const float v = pa[n * 32] * px[n * 32];
      if (v > t7) {                       // rare path (~3% of candidates)
        t7 = v;
        float hi, lo;
#define TK_SWAP(a, b) hi = fmaxf(a, b); lo = fminf(a, b); a = hi; b = lo;
        TK_SWAP(t6, t7); TK_SWAP(t5, t6); TK_SWAP(t4, t5); TK_SWAP(t3, t4);
        TK_SWAP(t2, t3); TK_SWAP(t1, t2); TK_SWAP(t0, t1);
#undef TK_SWAP
      }
    }
  }

  // ---- epilogue: out[m, j, f];  j=0 -> x[m,f], j=1..8 -> t[j-1]
  float vals[K_TOP + 1];
  vals[0] = x[(m0 + tx) * N_FEAT + (f0 + ty)];
  vals[1] = t0; vals[2] = t1; vals[3] = t2; vals[4] = t3;
  vals[5] = t4; vals[6] = t5; vals[7] = t6; vals[8] = t7;

#pragma unroll
  for (int j = 0; j < K_TOP + 1; ++j) {
    __syncthreads();
    s_o[tx][ty] = vals[j];               // stage (m=m0+tx, f=f0+ty)
    __syncthreads();
    // lanes sweep f -> 128B coalesced stores
    out[((size_t)(m0 + ty) * (K_TOP + 1) + j) * N_FEAT + f0 + tx] = s_o[ty][tx];
  }
}

extern "C" void kernel_launch(void* const* d_in, const int* in_sizes, int n_in,
                              void* d_out, int out_size, void* d_ws, size_t ws_size,
                              hipStream_t stream) {
  (void)in_sizes; (void)n_in; (void)out_size; (void)d_ws; (void)ws_size;
  const float* x   = (const float*)d_in[0];   // [2048, 64]
  const float* adj = (const float*)d_in[1];   // [2048, 2048]
  float* out       = (float*)d_out;           // [2048, 9, 64]

  dim3 block(32, 32);                 // 32 waves (wave32); lane = m_local
  dim3 grid(N_NODES / 32, N_FEAT / 32);
  topk_feat_kernel<<<grid, block, 0, stream>>>(x, adj, out);
}